// Dropout_16106127360680
// MI455X (gfx1250) — compile-verified
//
#include <hip/hip_runtime.h>
#include <hip/hip_bf16.h>

// Dropout (inverted, p = 0.5): out[i] = mask[i] ? x[i] * 2.0f : 0.0f
// Memory-bound streaming kernel for MI455X (gfx1250, wave32).
//
// Roofline: 9 B/element (4 load + 1 mask + 4 store) * 25.7M elements
//   ≈ 231 MB  ->  ~10 us at 23.3 TB/s HBM. Purely bandwidth-bound; no matrix
//   structure, so WMMA/TDM paths are inapplicable — the optimal path is wide
//   non-temporal VMEM (stream > 192 MB L2, zero reuse).
//
// Per thread (16 contiguous floats):
//   - mask: 1 x global_load_b128 th:TH_LOAD_NT   (16 bool bytes)
//   - x:    4 x global_load_b128 th:TH_LOAD_NT
//   - out:  4 x global_store_b128 th:TH_STORE_NT
// Mask bytes are 0/1 (jax bool), so the select is done arithmetically:
//   out = (x + x) * cvt_f32_ubyteN(maskword)
// -> v_pk_add_f32 + v_cvt_f32_ubyte{0..3} + v_dual_mul_f32, ~2.5 VALU/elem,
//    no per-byte and/cmp/cndmask chains.

typedef __attribute__((ext_vector_type(4))) float        f32x4;
typedef __attribute__((ext_vector_type(4))) unsigned int u32x4;

__global__ __launch_bounds__(256)
void dropout_vec16_kernel(const float* __restrict__ x,
                          const unsigned char* __restrict__ mask,
                          float* __restrict__ out,
                          long long n16)   // number of 16-element chunks
{
    long long i = (long long)blockIdx.x * blockDim.x + threadIdx.x;
    if (i >= n16) return;

    const long long base = i * 16;

    // 16 mask bytes in one 128-bit NT load; each u32 lane covers one f32x4.
    u32x4 m = __builtin_nontemporal_load((const u32x4*)(mask + base));

    const f32x4* __restrict__ xp = (const f32x4*)(x   + base);
    f32x4*       __restrict__ op = (f32x4*)      (out + base);

#pragma unroll
    for (int v = 0; v < 4; ++v) {
        f32x4 xv = __builtin_nontemporal_load(xp + v);
        unsigned int mw = m[v];

        // x * 2 as packed add (v_pk_add_f32 / v_dual_add_f32).
        f32x4 x2 = xv + xv;

        f32x4 ov;
        // (float)(byte k of mw) lowers to v_cvt_f32_ubyte{0,1,2,3} — a single
        // VALU op each, no byte-extract needed. mask byte is 0 or 1.
        ov[0] = x2[0] * (float)( mw         & 0xFFu);
        ov[1] = x2[1] * (float)((mw >>  8)  & 0xFFu);
        ov[2] = x2[2] * (float)((mw >> 16)  & 0xFFu);
        ov[3] = x2[3] * (float)( mw >> 24);

        __builtin_nontemporal_store(ov, op + v);
    }
}

// Scalar tail for n % 16 leftovers (zero for this problem size, kept so the
// launch is correct for arbitrary n).
__global__ __launch_bounds__(256)
void dropout_tail_kernel(const float* __restrict__ x,
                         const unsigned char* __restrict__ mask,
                         float* __restrict__ out,
                         long long start, long long n)
{
    long long i = start + (long long)blockIdx.x * blockDim.x + threadIdx.x;
    if (i < n) {
        out[i] = (x[i] + x[i]) * (float)mask[i];
    }
}

extern "C" void kernel_launch(void* const* d_in, const int* in_sizes, int n_in,
                              void* d_out, int out_size, void* d_ws, size_t ws_size,
                              hipStream_t stream) {
    const float*         x    = (const float*)d_in[0];
    const unsigned char* mask = (const unsigned char*)d_in[1];  // jax bool: 1 byte/elem
    float*               out  = (float*)d_out;

    const long long n   = (long long)in_sizes[0];   // 25,690,112
    const long long n16 = n / 16;
    const long long rem = n - n16 * 16;

    if (n16 > 0) {
        const int block = 256;  // 8 wave32 waves per block
        const long long grid = (n16 + block - 1) / block;
        dropout_vec16_kernel<<<(unsigned int)grid, block, 0, stream>>>(x, mask, out, n16);
    }
    if (rem > 0) {
        const int block = 256;
        const long long grid = (rem + block - 1) / block;
        dropout_tail_kernel<<<(unsigned int)grid, block, 0, stream>>>(x, mask, out,
                                                                      n16 * 16, n);
    }
}